// PlanarFlows_1546188227105
// MI455X (gfx1250) — compile-verified
//
#include <hip/hip_runtime.h>
#include <math.h>

// Planar flows: B=131072 rows, D=256, FLOWS=8.
// x_j = h0.w_j + b_j + sum_{i<j} elu(x_i) * T[i][j],   T[i][j] = u_hat_i . w_j
// h_out = h0 + sum_j elu(x_j) u_hat_j ;  logdet = sum_j log|1 + grad_j T[j][j]|

#define NB    131072
#define ND    256
#define NF    8

typedef __attribute__((ext_vector_type(2))) float v2f;
typedef __attribute__((ext_vector_type(4))) float v4f;
typedef __attribute__((ext_vector_type(8))) float v8f;

__device__ __forceinline__ float softplus_f(float x) {
    return fmaxf(x, 0.f) + log1pf(__expf(-fabsf(x)));
}

// ---------------------------------------------------------------------------
// Kernel 1: one block. Computes u_hat[8][256] and T[8][8] into workspace.
// ---------------------------------------------------------------------------
__global__ void pf_setup(const float* __restrict__ w, const float* __restrict__ u,
                         float* __restrict__ uhat, float* __restrict__ T) {
    __shared__ float s_uhat[NF * ND];
    const int warp = threadIdx.x >> 5;
    const int lane = threadIdx.x & 31;
    const float* wj = w + warp * ND;
    const float* uj = u + warp * ND;

    float wu = 0.f, nw = 0.f;
    for (int k = lane; k < ND; k += 32) {
        float wv = wj[k];
        wu += wv * uj[k];
        nw += wv * wv;
    }
    for (int s = 16; s > 0; s >>= 1) {
        wu += __shfl_xor(wu, s, 32);
        nw += __shfl_xor(nw, s, 32);
    }
    const float coef  = -1.f + softplus_f(wu) - wu;
    const float scale = coef / nw;
    for (int k = lane; k < ND; k += 32) {
        float uh = uj[k] + scale * wj[k];
        s_uhat[warp * ND + k] = uh;
        uhat[warp * ND + k]   = uh;
    }
    __syncthreads();

    // T[i][warp] = u_hat_i . w_warp
    for (int i = 0; i < NF; ++i) {
        float t = 0.f;
        for (int k = lane; k < ND; k += 32)
            t += s_uhat[i * ND + k] * wj[k];
        for (int s = 16; s > 0; s >>= 1)
            t += __shfl_xor(t, s, 32);
        if (lane == 0) T[i * NF + warp] = t;
    }
}

// ---------------------------------------------------------------------------
// Kernel 2: main. 256 threads = 8 waves; each wave owns a 16-row tile.
// Pass 1: X0 = W_pad @ h_tile^T via v_wmma_f32_16x16x4_f32 (flow in VGPR dim,
//         batch-row in lane dim -> recurrence is lane-local, no transpose).
// Pass 2: h_out = h0 + E @ u_hat  (coalesced b128, u_hat broadcast from LDS).
// ---------------------------------------------------------------------------
__global__ void __launch_bounds__(256)
pf_main(const float* __restrict__ h0, const float* __restrict__ w,
        const float* __restrict__ bflow, const float* __restrict__ uhat,
        const float* __restrict__ T, float* __restrict__ hout,
        float* __restrict__ logd) {
    __shared__ float s_uhat[NF * ND];    // 8 KB
    __shared__ float s_wpad[16 * ND];    // 16 KB, flows 8..15 zeroed

    const int tid = threadIdx.x;
    for (int i = tid; i < NF * ND; i += 256) {
        s_uhat[i]            = uhat[i];
        s_wpad[i]            = w[i];
        s_wpad[NF * ND + i]  = 0.f;
    }
    __syncthreads();

    const int wave = tid >> 5;
    const int lane = tid & 31;
    const int m    = lane & 15;   // A: flow row (pad>=8). B/C: batch row in tile.
    const int hi   = lane >> 4;   // K-pair selector within 16x16x4 slice.
    const int row0 = blockIdx.x * 128 + wave * 16;

    const float* hrow = h0 + (size_t)(row0 + m) * ND;

    // ---- Pass 1: 64 WMMAs over K=256, two accumulators to break RAW chain.
    v8f acc0 = {0.f, 0.f, 0.f, 0.f, 0.f, 0.f, 0.f, 0.f};
    v8f acc1 = {0.f, 0.f, 0.f, 0.f, 0.f, 0.f, 0.f, 0.f};
#pragma unroll 4
    for (int ks = 0; ks < 64; ks += 2) {
        const int k0 = ks * 4 + (hi << 1);
        v2f a0 = *(const v2f*)&s_wpad[m * ND + k0];
        v2f b0 = *(const v2f*)&hrow[k0];
        acc0 = __builtin_amdgcn_wmma_f32_16x16x4_f32(
            false, a0, false, b0, (short)0, acc0, false, false);
        const int k1 = k0 + 4;
        v2f a1 = *(const v2f*)&s_wpad[m * ND + k1];
        v2f b1 = *(const v2f*)&hrow[k1];
        acc1 = __builtin_amdgcn_wmma_f32_16x16x4_f32(
            false, a1, false, b1, (short)0, acc1, false, false);
    }
    v8f xc = acc0 + acc1;   // lane L<16: xc[j] = h0[row0+L] . w_j

    // ---- Per-row recurrence (lane-local; lanes 16..31 compute unused values).
    float e[NF];
    float ld = 0.f;
#pragma unroll
    for (int j = 0; j < NF; ++j) {
        float x = xc[j] + bflow[j];
#pragma unroll
        for (int i = 0; i < NF; ++i)
            if (i < j) x += e[i] * T[i * NF + j];
        float ex = __expf(x);
        float g  = x > 0.f ? 1.f : ex;              // d/dx elu
        ld += __logf(fabsf(1.f + g * T[j * NF + j]));
        e[j] = x > 0.f ? x : ex - 1.f;              // elu(x)
    }
    if (hi == 0) logd[row0 + m] = ld;

    // ---- Pass 2: h_out = h0 + E @ u_hat, coalesced 2-rows-per-iteration.
#pragma unroll
    for (int mm = 0; mm < 16; mm += 2) {
        const int r = mm + hi;   // lanes 0..15 -> row mm, lanes 16..31 -> mm+1
        float er[NF];
#pragma unroll
        for (int j = 0; j < NF; ++j) er[j] = __shfl(e[j], r, 32);
        const float* hr = h0   + (size_t)(row0 + r) * ND;
        float*       ho = hout + (size_t)(row0 + r) * ND;
#pragma unroll
        for (int cc = 0; cc < 4; ++cc) {
            const int col = cc * 64 + m * 4;
            v4f v = *(const v4f*)&hr[col];
#pragma unroll
            for (int j = 0; j < NF; ++j) {
                v4f uv = *(const v4f*)&s_uhat[j * ND + col];
                v = v + er[j] * uv;
            }
            *(v4f*)&ho[col] = v;
        }
    }
}

// ---------------------------------------------------------------------------
extern "C" void kernel_launch(void* const* d_in, const int* in_sizes, int n_in,
                              void* d_out, int out_size, void* d_ws, size_t ws_size,
                              hipStream_t stream) {
    const float* h0 = (const float*)d_in[0];   // [131072, 256]
    const float* w  = (const float*)d_in[1];   // [8, 256]
    const float* b  = (const float*)d_in[2];   // [8]
    const float* u  = (const float*)d_in[3];   // [8, 256]

    float* hout = (float*)d_out;                           // [131072, 256]
    float* logd = (float*)d_out + (size_t)NB * ND;         // [131072]

    float* uhat = (float*)d_ws;            // 8*256 floats
    float* T    = (float*)d_ws + NF * ND;  // 8*8 floats

    pf_setup<<<dim3(1), dim3(256), 0, stream>>>(w, u, uhat, T);
    pf_main<<<dim3(NB / 128), dim3(256), 0, stream>>>(h0, w, b, uhat, T, hout, logd);
}